// CopyOnlyGenerator_85014582657768
// MI455X (gfx1250) — compile-verified
//
#include <hip/hip_runtime.h>
#include <hip/hip_bf16.h>

typedef __attribute__((ext_vector_type(2))) float v2f;
typedef __attribute__((ext_vector_type(8))) float v8f;

#define B_   32
#define T_   512
#define S_   512
#define H_   1024
#define DVS_ 2048
#define DVT_ 1024
#define VOUT_ (1 + DVS_ + DVT_)   // 3073

// ---------------------------------------------------------------------------
// Kernel 1: logits = hiddens(16xH tile) * W^T (cols 0..2 valid) via
// V_WMMA_F32_16X16X4_F32, then 3-way softmax -> p (B*T, 3).
// Block = 128 threads = 4 waves; wave w covers K in [w*H/4, (w+1)*H/4),
// partial accumulators reduced through LDS. EXEC stays all-1s in the loop
// (branchless cndmask masking of the B fragment, no divergent loads).
//
// Per 8-wide K chunk each lane loads 16 contiguous bytes (b128) for A and B;
// K is permuted identically on both operands (dot products are order
// invariant): WMMA#1 <- actual K {k,k+1} (half 0) / {k+4,k+5} (half 1),
// WMMA#2 <- {k+2,k+3} / {k+6,k+7}.
//
// A layout (16x4 f32): lanes 0-15 hold M=lane, K=0..1; lanes 16-31 M=lane-16, K=2..3.
// B layout (4x16 f32): lanes 0-15 hold N=lane, K=0..1; lanes 16-31 N=lane-16, K=2..3.
// C/D layout: VGPR i -> M=i (lanes 0-15, N=lane) / M=8+i (lanes 16-31, N=lane-16).
// ---------------------------------------------------------------------------
__global__ __launch_bounds__(128) void logits_softmax_wmma(
    const float* __restrict__ hiddens,  // (B*T, H)
    const float* __restrict__ W,        // (3, H)
    const float* __restrict__ bvec,     // (3,)
    float* __restrict__ p_out)          // (B*T, 3)
{
  const int tid  = threadIdx.x;
  const int lane = tid & 31;
  const int wave = tid >> 5;          // 0..3 : K-split
  const int row0 = blockIdx.x * 16;   // tile of 16 rows of (B*T)
  const int n    = lane & 15;         // M for A, N for B/C
  const int half = lane >> 4;         // 0 or 1

  const float* __restrict__ arow = hiddens + (size_t)(row0 + n) * H_;
  const float* __restrict__ wrow = W + (size_t)(n < 3 ? n : 0) * H_;  // clamped, always valid
  const bool wvalid = (n < 3);

  const int k0 = wave * (H_ / 4);
  const int k1 = k0 + (H_ / 4);

  __builtin_prefetch(arow + k0, 0, 0);   // global_prefetch_b8 on the A stream

  v8f c = {0.f, 0.f, 0.f, 0.f, 0.f, 0.f, 0.f, 0.f};

  for (int k = k0; k < k1; k += 8) {
    const float4 af = *(const float4*)(arow + k + 4 * half);  // global_load_b128
    const float4 wf = *(const float4*)(wrow + k + 4 * half);  // global_load_b128
    v2f a1, a2, b1, b2;
    a1.x = af.x;  a1.y = af.y;
    a2.x = af.z;  a2.y = af.w;
    b1.x = wvalid ? wf.x : 0.f;  b1.y = wvalid ? wf.y : 0.f;  // v_cndmask, no branch
    b2.x = wvalid ? wf.z : 0.f;  b2.y = wvalid ? wf.w : 0.f;
    // 8 args: (neg_a, A, neg_b, B, c_mod, C, reuse_a, reuse_b)
    c = __builtin_amdgcn_wmma_f32_16x16x4_f32(
        false, a1, false, b1, (short)0, c, false, false);
    c = __builtin_amdgcn_wmma_f32_16x16x4_f32(
        false, a2, false, b2, (short)0, c, false, false);
  }

  __shared__ float lg[4][16 * 3];
  if (wvalid) {
    #pragma unroll
    for (int i = 0; i < 8; ++i) {
      const int m = i + 8 * half;     // row within tile
      lg[wave][m * 3 + n] = c[i];
    }
  }
  __syncthreads();

  if (tid < 16) {
    const int m = tid;
    float l0 = bvec[0], l1 = bvec[1], l2 = bvec[2];
    #pragma unroll
    for (int w2 = 0; w2 < 4; ++w2) {
      l0 += lg[w2][m * 3 + 0];
      l1 += lg[w2][m * 3 + 1];
      l2 += lg[w2][m * 3 + 2];
    }
    const float mx = fmaxf(l0, fmaxf(l1, l2));
    const float e0 = expf(l0 - mx);
    const float e1 = expf(l1 - mx);
    const float e2 = expf(l2 - mx);
    const float inv = 1.0f / (e0 + e1 + e2);
    const size_t r = (size_t)(row0 + m) * 3;
    p_out[r + 0] = e0 * inv;
    p_out[r + 1] = e1 * inv;
    p_out[r + 2] = e2 * inv;
  }
}

// ---------------------------------------------------------------------------
// Kernel 2: one-hot rows -> integer index. One wave per row, float4 scans.
// ---------------------------------------------------------------------------
__global__ __launch_bounds__(256) void onehot_to_idx(
    const float* __restrict__ maps, int V, int* __restrict__ idx_out)
{
  const int wave = (int)((blockIdx.x * blockDim.x + threadIdx.x) >> 5);
  const int lane = threadIdx.x & 31;
  const float* __restrict__ rowp = maps + (size_t)wave * V;

  int found = 0;                      // store idx+1; one-hot guarantees a hit
  for (int v = lane * 4; v < V; v += 32 * 4) {
    const float4 x = *(const float4*)(rowp + v);
    if (x.x > 0.5f) found = v + 1;
    if (x.y > 0.5f) found = v + 2;
    if (x.z > 0.5f) found = v + 3;
    if (x.w > 0.5f) found = v + 4;
  }
  if (found) idx_out[wave] = found - 1;
}

// ---------------------------------------------------------------------------
// Kernel 3: per-(b,t) output row assembly in LDS (scatter with ds_add_f32),
// coalesced row store, block argmax (first-max wins, matching jnp.argmax).
// ---------------------------------------------------------------------------
__global__ __launch_bounds__(256) void assemble_row(
    const float* __restrict__ src_attn,  // (B,T,S)
    const float* __restrict__ tgt_attn,  // (B,T,T)
    const int*   __restrict__ src_idx,   // (B,S)
    const int*   __restrict__ tgt_idx,   // (B,T)
    const float* __restrict__ p,         // (B*T, 3)
    float* __restrict__ probs_out,       // (B*T, VOUT_)
    float* __restrict__ pred_out)        // (B*T)
{
  const int bt  = blockIdx.x;           // 0..B*T-1
  const int b   = bt / T_;
  const int tid = threadIdx.x;

  __shared__ float rowv[VOUT_];
  __shared__ float rmax[256];
  __shared__ int   ridx[256];

  for (int i = tid; i < VOUT_; i += 256) rowv[i] = 0.f;
  __syncthreads();

  const float p0 = p[(size_t)bt * 3 + 0];
  const float p1 = p[(size_t)bt * 3 + 1];
  const float p2 = p[(size_t)bt * 3 + 2];

  const float* __restrict__ sa = src_attn + (size_t)bt * S_;
  const int*   __restrict__ si = src_idx + (size_t)b * S_;
  for (int s = tid; s < S_; s += 256) {
    atomicAdd(&rowv[1 + si[s]], p0 * sa[s]);            // ds_add_f32
  }

  const float* __restrict__ ta = tgt_attn + (size_t)bt * T_;
  const int*   __restrict__ ti = tgt_idx + (size_t)b * T_;
  for (int t = tid; t < T_; t += 256) {
    atomicAdd(&rowv[1 + DVS_ + ti[t]], p1 * ta[t]);     // ds_add_f32
  }

  if (tid == 0) rowv[0] = p2;
  __syncthreads();

  float best = -1.0f;
  int bidx = VOUT_;
  float* __restrict__ out = probs_out + (size_t)bt * VOUT_;
  for (int i = tid; i < VOUT_; i += 256) {
    const float v = rowv[i];
    out[i] = v;
    if (v > best) { best = v; bidx = i; }               // keeps lowest i per thread
  }

  rmax[tid] = best;
  ridx[tid] = bidx;
  __syncthreads();
  for (int off = 128; off > 0; off >>= 1) {
    if (tid < off) {
      const float o = rmax[tid + off];
      const int  oi = ridx[tid + off];
      if (o > rmax[tid] || (o == rmax[tid] && oi < ridx[tid])) {
        rmax[tid] = o;
        ridx[tid] = oi;
      }
    }
    __syncthreads();
  }
  if (tid == 0) pred_out[bt] = (float)ridx[0];
}

// ---------------------------------------------------------------------------
// Launch wrapper
// Inputs (setup_inputs order):
//  0 hiddens (B,T,H) f32            1 source_attentions (B,T,S) f32
//  2 source_attention_maps (B,S,DVS) f32
//  3 target_attentions (B,T,T) f32  4 target_attention_maps (B,T,DVT) f32
//  5 W (3,H) f32                    6 b (3,) f32
// Output: probs (B,T,VOUT_) f32 then predictions (B,T) as f32.
// ---------------------------------------------------------------------------
extern "C" void kernel_launch(void* const* d_in, const int* in_sizes, int n_in,
                              void* d_out, int out_size, void* d_ws, size_t ws_size,
                              hipStream_t stream) {
  const float* hiddens  = (const float*)d_in[0];
  const float* src_attn = (const float*)d_in[1];
  const float* src_maps = (const float*)d_in[2];
  const float* tgt_attn = (const float*)d_in[3];
  const float* tgt_maps = (const float*)d_in[4];
  const float* W        = (const float*)d_in[5];
  const float* bvec     = (const float*)d_in[6];

  const int BT = B_ * T_;

  // Workspace layout
  char* ws = (char*)d_ws;
  float* p_ws    = (float*)ws;                               // BT*3 floats
  int*   sidx_ws = (int*)(ws + (size_t)BT * 3 * sizeof(float));
  int*   tidx_ws = (int*)((char*)sidx_ws + (size_t)B_ * S_ * sizeof(int));

  float* probs_out = (float*)d_out;                          // BT * VOUT_
  float* pred_out  = probs_out + (size_t)BT * VOUT_;         // BT

  // 1) p = softmax(hiddens @ W^T + b) via fp32 WMMA (4-way K-split per block)
  logits_softmax_wmma<<<BT / 16, 128, 0, stream>>>(hiddens, W, bvec, p_ws);

  // 2) recover indices from one-hot maps (one wave per row, 8 waves/block)
  onehot_to_idx<<<(B_ * S_) / 8, 256, 0, stream>>>(src_maps, DVS_, sidx_ws);
  onehot_to_idx<<<(B_ * T_) / 8, 256, 0, stream>>>(tgt_maps, DVT_, tidx_ws);

  // 3) scatter-assemble output rows + argmax
  assemble_row<<<BT, 256, 0, stream>>>(src_attn, tgt_attn, sidx_ws, tidx_ws,
                                       p_ws, probs_out, pred_out);
}